// KeGNN_2147483648537
// MI455X (gfx1250) — compile-verified
//
#include <hip/hip_runtime.h>
#include <cmath>

// Problem constants (from reference)
constexpr int IN_DIM = 128;
constexpr int HID    = 64;
constexpr int OUTC   = 40;
constexpr int OUTP   = 48;   // OUT padded to 3 WMMA n-tiles

constexpr int XPAD   = 4;                 // LDS row pad (DWORDs) to break bank conflicts
constexpr int XLD    = IN_DIM + XPAD;     // 132 floats per staged x row
constexpr int HLD    = HID + XPAD;        // 68 floats per staged h1 row

typedef __attribute__((ext_vector_type(2))) float v2f;
typedef __attribute__((ext_vector_type(8))) float v8f;
typedef __attribute__((ext_vector_type(4))) unsigned v4u;
typedef __attribute__((ext_vector_type(4))) int v4i;
typedef __attribute__((ext_vector_type(8))) int v8i;

#if __has_builtin(__builtin_amdgcn_tensor_load_to_lds) && __has_builtin(__builtin_amdgcn_s_wait_tensorcnt)
#define USE_TDM 1
#else
#define USE_TDM 0
#endif

#if USE_TDM
// Issue a 2-D TDM tile load: rows x cols elements of f32, row-major global with
// row stride = cols, into LDS at ldsOff with +4 DWORDs pad per row.
// padIntervalCode: pad after 2^(code+1) DWORDs stored (code 6 -> 128 DWORDs, code 5 -> 64).
__device__ __forceinline__ void tdm_load_tile(const float* gsrc, unsigned ldsOff,
                                              unsigned cols, unsigned rowsAvail,
                                              unsigned tileRows, unsigned padIntervalCode) {
    unsigned long long ga = (unsigned long long)(uintptr_t)gsrc;
    v4u g0;
    g0.x = 1u;                                               // count=1 (valid), load, no gather
    g0.y = ldsOff;                                           // lds_addr
    g0.z = (unsigned)(ga & 0xFFFFFFFFull);                   // global_addr[31:0]
    g0.w = (unsigned)((ga >> 32) & 0x01FFFFFFull) | (2u << 30); // addr[56:32] | type=2
    const unsigned padAmtCode = 3u;                          // 4 DWORDs of pad per row
    unsigned w0 = (2u << 16)                                  // data_size = 4 bytes
                | (1u << 20)                                  // pad_enable
                | (padIntervalCode << 22)
                | (padAmtCode << 25);
    unsigned w1 = (cols & 0xFFFFu) << 16;                     // tensor_dim0 lo (barrier addr = 0)
    unsigned w2 = (cols >> 16) | ((rowsAvail & 0xFFFFu) << 16); // dim0 hi | dim1 lo
    unsigned w3 = (rowsAvail >> 16) | (cols << 16);           // dim1 hi | tile_dim0
    unsigned w4 = tileRows;                                   // tile_dim1, tile_dim2=0
    unsigned w5 = cols;                                       // tensor_dim0_stride[31:0]
    unsigned w6 = 0u;                                         // stride0[47:32]=0, stride1 lo=0
    unsigned w7 = 0u;
    v8i g1  = { (int)w0, (int)w1, (int)w2, (int)w3, (int)w4, (int)w5, (int)w6, (int)w7 };
    v4i gz4 = { 0, 0, 0, 0 };                                 // 2-D tensor: groups 2/3 unused
    v8i gz8 = { 0, 0, 0, 0, 0, 0, 0, 0 };
    __builtin_amdgcn_tensor_load_to_lds(g0, g1, gz4, gz4, gz8, 0);
}
#endif

// ---------------------------------------------------------------- degree prep
__global__ void set_deg_kernel(float* __restrict__ deg, int n) {
    int i = blockIdx.x * blockDim.x + threadIdx.x;
    if (i < n) deg[i] = 1.0f;                       // self-loop contribution
}

__global__ void accum_deg_kernel(const int* __restrict__ dst, float* __restrict__ deg, int e) {
    int i = blockIdx.x * blockDim.x + threadIdx.x;
    if (i < e) atomicAdd(&deg[dst[i]], 1.0f);
}

__global__ void rsqrt_inplace_kernel(float* __restrict__ d, int n) {
    int i = blockIdx.x * blockDim.x + threadIdx.x;
    if (i < n) d[i] = rsqrtf(d[i]);
}

// ------------------------------------------------------- GEMM1: hw1 = x @ W1
// TDM stages the 64x128 activation tile into LDS (padded rows), WMMA f32 16x16x4
// computes, epilogue fuses self-loop init: agg1 = dinv[r]^2 * hw1[r].
__global__ __launch_bounds__(128)
void gemm1_kernel(const float* __restrict__ x, const float* __restrict__ W,
                  const float* __restrict__ dinv,
                  float* __restrict__ hw, float* __restrict__ agg, int n) {
    __shared__ float sW[IN_DIM * HID];              // 32 KB, [K][N]
    __shared__ float sX[64 * XLD];                  // 33 KB, padded x tile

    const int blockRow = blockIdx.x * 64;

#if USE_TDM
    if (threadIdx.x < 32) {                         // wave 0 issues the DMA
        tdm_load_tile(x + (size_t)blockRow * IN_DIM,
                      (unsigned)(uintptr_t)(void*)sX,
                      IN_DIM, (unsigned)(n - blockRow), 64, /*128 DW interval*/ 6u);
    }
#endif
    for (int t = threadIdx.x; t < IN_DIM * HID; t += 128) sW[t] = W[t];
#if !USE_TDM
    for (int t = threadIdx.x; t < 64 * IN_DIM; t += 128) {
        int r = t >> 7, c = t & (IN_DIM - 1);
        int gr = blockRow + r;
        sX[r * XLD + c] = (gr < n) ? x[(size_t)gr * IN_DIM + c] : 0.0f;
    }
#else
    if (threadIdx.x < 32) __builtin_amdgcn_s_wait_tensorcnt(0);
#endif
    __syncthreads();

    const int lane = threadIdx.x & 31;
    const int wv   = threadIdx.x >> 5;
    const int half = lane >> 4;                     // K-half select (0/1)
    const int m    = lane & 15;
    const int rowBase = blockRow + wv * 16;
    if (rowBase >= n) return;

    const float* xrow = sX + (wv * 16 + m) * XLD + 2 * half;

    v8f acc0 = {}, acc1 = {}, acc2 = {}, acc3 = {};
    #pragma unroll
    for (int k0 = 0; k0 < IN_DIM; k0 += 4) {
        v2f a;                                      // A 16x4: lane row, K = k0+2*half+{0,1}
        a.x = xrow[k0];
        a.y = xrow[k0 + 1];
        const float* bp = &sW[(k0 + 2 * half) * HID + m];
        v2f b0, b1, b2, b3;                         // B 4x16: K rows striped across lanes
        b0.x = bp[0];  b0.y = bp[HID + 0];
        b1.x = bp[16]; b1.y = bp[HID + 16];
        b2.x = bp[32]; b2.y = bp[HID + 32];
        b3.x = bp[48]; b3.y = bp[HID + 48];
        acc0 = __builtin_amdgcn_wmma_f32_16x16x4_f32(false, a, false, b0, (short)0, acc0, false, false);
        acc1 = __builtin_amdgcn_wmma_f32_16x16x4_f32(false, a, false, b1, (short)0, acc1, false, false);
        acc2 = __builtin_amdgcn_wmma_f32_16x16x4_f32(false, a, false, b2, (short)0, acc2, false, false);
        acc3 = __builtin_amdgcn_wmma_f32_16x16x4_f32(false, a, false, b3, (short)0, acc3, false, false);
    }

    #pragma unroll
    for (int j = 0; j < 8; ++j) {                   // D: VGPR j -> M = j + 8*half, N = m
        const int r = rowBase + j + 8 * half;
        if (r < n) {
            const float di = dinv[r];
            const float w  = di * di;               // self-loop norm
            float* hr = hw  + (size_t)r * HID + m;
            float* ar = agg + (size_t)r * HID + m;
            hr[0]  = acc0[j];  ar[0]  = w * acc0[j];
            hr[16] = acc1[j];  ar[16] = w * acc1[j];
            hr[32] = acc2[j];  ar[32] = w * acc2[j];
            hr[48] = acc3[j];  ar[48] = w * acc3[j];
        }
    }
}

// ------------------------------------------------- GEMM2: hw2 = h1 @ W2 (pad 48)
__global__ __launch_bounds__(128)
void gemm2_kernel(const float* __restrict__ h1, const float* __restrict__ W,
                  const float* __restrict__ dinv,
                  float* __restrict__ hw, float* __restrict__ agg, int n) {
    __shared__ float sW[HID * OUTP];                // 12 KB, [K][Npad], zero-padded cols
    __shared__ float sH[64 * HLD];                  // 17 KB, padded h1 tile

    const int blockRow = blockIdx.x * 64;

#if USE_TDM
    if (threadIdx.x < 32) {
        tdm_load_tile(h1 + (size_t)blockRow * HID,
                      (unsigned)(uintptr_t)(void*)sH,
                      HID, (unsigned)(n - blockRow), 64, /*64 DW interval*/ 5u);
    }
#endif
    for (int t = threadIdx.x; t < HID * OUTP; t += 128) {
        int k = t / OUTP, c = t % OUTP;
        sW[t] = (c < OUTC) ? W[k * OUTC + c] : 0.0f;
    }
#if !USE_TDM
    for (int t = threadIdx.x; t < 64 * HID; t += 128) {
        int r = t >> 6, c = t & (HID - 1);
        int gr = blockRow + r;
        sH[r * HLD + c] = (gr < n) ? h1[(size_t)gr * HID + c] : 0.0f;
    }
#else
    if (threadIdx.x < 32) __builtin_amdgcn_s_wait_tensorcnt(0);
#endif
    __syncthreads();

    const int lane = threadIdx.x & 31;
    const int wv   = threadIdx.x >> 5;
    const int half = lane >> 4;
    const int m    = lane & 15;
    const int rowBase = blockRow + wv * 16;
    if (rowBase >= n) return;

    const float* hrow = sH + (wv * 16 + m) * HLD + 2 * half;

    v8f acc0 = {}, acc1 = {}, acc2 = {};
    #pragma unroll
    for (int k0 = 0; k0 < HID; k0 += 4) {
        v2f a;
        a.x = hrow[k0];
        a.y = hrow[k0 + 1];
        const float* bp = &sW[(k0 + 2 * half) * OUTP + m];
        v2f b0, b1, b2;
        b0.x = bp[0];  b0.y = bp[OUTP + 0];
        b1.x = bp[16]; b1.y = bp[OUTP + 16];
        b2.x = bp[32]; b2.y = bp[OUTP + 32];
        acc0 = __builtin_amdgcn_wmma_f32_16x16x4_f32(false, a, false, b0, (short)0, acc0, false, false);
        acc1 = __builtin_amdgcn_wmma_f32_16x16x4_f32(false, a, false, b1, (short)0, acc1, false, false);
        acc2 = __builtin_amdgcn_wmma_f32_16x16x4_f32(false, a, false, b2, (short)0, acc2, false, false);
    }

    #pragma unroll
    for (int j = 0; j < 8; ++j) {
        const int r = rowBase + j + 8 * half;
        if (r < n) {
            const float di = dinv[r];
            const float w  = di * di;
            float* hr = hw  + (size_t)r * OUTP + m;
            float* ar = agg + (size_t)r * OUTP + m;
            hr[0]  = acc0[j];  ar[0]  = w * acc0[j];
            hr[16] = acc1[j];  ar[16] = w * acc1[j];
            hr[32] = acc2[j];  ar[32] = w * acc2[j];
        }
    }
}

// ------------------------------------------ edge scatter: agg[dst] += norm*hw[src]
// One wave32 per edge; lanes cover the feature row (coalesced, L2-resident).
template <int F>
__global__ void scatter_kernel(const int* __restrict__ src, const int* __restrict__ dst,
                               const float* __restrict__ dinv,
                               const float* __restrict__ hw, float* __restrict__ agg, int e) {
    int gid  = blockIdx.x * blockDim.x + threadIdx.x;
    int ew   = gid >> 5;
    int lane = gid & 31;
    if (ew >= e) return;
    const int s = src[ew];
    const int d = dst[ew];
    const float w = dinv[s] * dinv[d];
    const float* hs = hw  + (size_t)s * F;
    float*       ad = agg + (size_t)d * F;
    #pragma unroll
    for (int f = lane; f < F; f += 32)
        atomicAdd(&ad[f], w * hs[f]);
}

// -------------------------------------------------------- h1 = relu(agg1 + b1)
__global__ void relu_bias_kernel(float* __restrict__ a, const float* __restrict__ b, int total) {
    int i = blockIdx.x * blockDim.x + threadIdx.x;
    if (i < total) {
        int c = i & (HID - 1);
        a[i] = fmaxf(a[i] + b[c], 0.0f);
    }
}

// ------------------- y = relu(agg2 + b2 + P*K*U); out = log_softmax(y, axis=1)
__global__ void finalize_kernel(const float* __restrict__ agg2, const float* __restrict__ b2,
                                const float* __restrict__ P, const float* __restrict__ K,
                                const float* __restrict__ U, float* __restrict__ out, int n) {
    int r = blockIdx.x * blockDim.x + threadIdx.x;
    if (r >= n) return;
    float y[OUTC];
    float mx = -INFINITY;
    #pragma unroll
    for (int c = 0; c < OUTC; ++c) {
        float v = agg2[(size_t)r * OUTP + c] + b2[c] + P[c] * K[c] * U[c];
        v = fmaxf(v, 0.0f);
        y[c] = v;
        mx = fmaxf(mx, v);
    }
    float sum = 0.0f;
    #pragma unroll
    for (int c = 0; c < OUTC; ++c) sum += __expf(y[c] - mx);
    const float lse = mx + __logf(sum);
    #pragma unroll
    for (int c = 0; c < OUTC; ++c) out[(size_t)r * OUTC + c] = y[c] - lse;
}

// --------------------------------------------------------------------- launch
extern "C" void kernel_launch(void* const* d_in, const int* in_sizes, int n_in,
                              void* d_out, int out_size, void* d_ws, size_t ws_size,
                              hipStream_t stream) {
    const float* x  = (const float*)d_in[0];
    const int*   ei = (const int*)d_in[1];
    const float* W1 = (const float*)d_in[2];
    const float* b1 = (const float*)d_in[3];
    const float* W2 = (const float*)d_in[4];
    const float* b2 = (const float*)d_in[5];
    const float* P  = (const float*)d_in[6];
    const float* K  = (const float*)d_in[7];
    const float* U  = (const float*)d_in[8];
    float* out = (float*)d_out;

    const int n = in_sizes[0] / IN_DIM;     // 100000
    const int e = in_sizes[1] / 2;          // 1600000
    const int* srcI = ei;                   // edge_index[0]
    const int* dstI = ei + e;               // edge_index[1]

    // Workspace layout (floats), agg2 aliases hw1 (dead after scatter1).
    float* ws = (float*)d_ws;
    size_t o = 0;
    float* deg  = ws + o; o += ((size_t)n + 255) & ~(size_t)255;
    float* hw1  = ws + o; o += ((size_t)n * HID + 255) & ~(size_t)255;
    float* agg1 = ws + o; o += ((size_t)n * HID + 255) & ~(size_t)255;
    float* hw2  = ws + o; o += ((size_t)n * OUTP + 255) & ~(size_t)255;
    float* agg2 = hw1;
    (void)ws_size; (void)n_in; (void)out_size;

    const int B = 256;
    // degree (self-loops folded in) and dinv
    set_deg_kernel<<<(n + B - 1) / B, B, 0, stream>>>(deg, n);
    accum_deg_kernel<<<(e + B - 1) / B, B, 0, stream>>>(dstI, deg, e);
    rsqrt_inplace_kernel<<<(n + B - 1) / B, B, 0, stream>>>(deg, n);
    const float* dinv = deg;

    // layer 1
    gemm1_kernel<<<(n + 63) / 64, 128, 0, stream>>>(x, W1, dinv, hw1, agg1, n);
    {
        long long threads = (long long)e * 32;
        scatter_kernel<HID><<<(unsigned)((threads + B - 1) / B), B, 0, stream>>>(srcI, dstI, dinv, hw1, agg1, e);
    }
    relu_bias_kernel<<<((long long)n * HID + B - 1) / B, B, 0, stream>>>(agg1, b1, n * HID);

    // layer 2 (agg1 now holds h1)
    gemm2_kernel<<<(n + 63) / 64, 128, 0, stream>>>(agg1, W2, dinv, hw2, agg2, n);
    {
        long long threads = (long long)e * 32;
        scatter_kernel<OUTP><<<(unsigned)((threads + B - 1) / B), B, 0, stream>>>(srcI, dstI, dinv, hw2, agg2, e);
    }

    // knowledge layer + log_softmax
    finalize_kernel<<<(n + 127) / 128, 128, 0, stream>>>(agg2, b2, P, K, U, out, n);
}